// DLASSO_GNNHyp_10677288698539
// MI455X (gfx1250) — compile-verified
//
#include <hip/hip_runtime.h>
#include <math.h>

// Problem constants (from reference)
#define B_   16
#define P_   50
#define M_   512
#define N_   1024
#define KST  20     // k_steps
#define E_   400    // edges per batch (2*EH)

typedef __attribute__((ext_vector_type(2))) float v2f;
typedef __attribute__((ext_vector_type(8))) float v8f;

#if defined(__gfx1250__) && __has_builtin(__builtin_amdgcn_wmma_f32_16x16x4_f32)
#define HAVE_WMMA_F32 1
#endif

// D(16x16,f32) += A(16x4,f32) x B(4x16,f32), full-precision CDNA5 WMMA.
// Per-lane operand layout (ISA 7.12.2): row/col = lane%16, K = 2*(lane/16)+v.
__device__ __forceinline__ v8f wmma4(v2f a, v2f b, v8f c) {
#ifdef HAVE_WMMA_F32
  return __builtin_amdgcn_wmma_f32_16x16x4_f32(
      /*neg_a=*/false, a, /*neg_b=*/false, b,
      /*c_mod=*/(short)0, c, /*reuse_a=*/false, /*reuse_b=*/false);
#else
  // Layout-faithful VALU fallback (compile safety only).
  v8f d = c;
  int lane = threadIdx.x & 31;
  int col = lane & 15, hi = lane >> 4;
#pragma unroll
  for (int r = 0; r < 8; ++r) {
    int row = r + 8 * hi;
    float acc = 0.f;
#pragma unroll
    for (int k = 0; k < 4; ++k) {
      float av = __shfl((k & 1) ? a.y : a.x, row + 16 * (k >> 1), 32);
      float bv = __shfl((k & 1) ? b.y : b.x, col + 16 * (k >> 1), 32);
      acc += av * bv;
    }
    d[r] += acc;
  }
  return d;
#endif
}

// ---- Generic 16-row GEMM: out(16,N) = X(16,K) @ W(K,N) + bias ---------------
// One wave per 64-column block (4x 16-col tiles); A operand (X rows) loaded
// once per k-step and reused by all 4 WMMAs.
__global__ void k_gemm16(const float* __restrict__ X, const float* __restrict__ W,
                         const float* __restrict__ bias, float* __restrict__ out,
                         int K, int N, int act) {
  int tile0 = blockIdx.x * 64;
  int lane = threadIdx.x & 31;
  int rw = lane & 15, hi = lane >> 4;
  v8f acc0 = {}, acc1 = {}, acc2 = {}, acc3 = {};
  const float* xrow = X + (size_t)rw * K;
  for (int kb = 0; kb < K; kb += 4) {
    int k0 = kb + 2 * hi;
    v2f a; a.x = xrow[k0]; a.y = xrow[k0 + 1];
    const float* w0 = W + (size_t)k0 * N + tile0 + rw;
    const float* w1 = W + (size_t)(k0 + 1) * N + tile0 + rw;
    v2f b;
    b.x = w0[0];  b.y = w1[0];  acc0 = wmma4(a, b, acc0);
    b.x = w0[16]; b.y = w1[16]; acc1 = wmma4(a, b, acc1);
    b.x = w0[32]; b.y = w1[32]; acc2 = wmma4(a, b, acc2);
    b.x = w0[48]; b.y = w1[48]; acc3 = wmma4(a, b, acc3);
  }
#pragma unroll
  for (int tq = 0; tq < 4; ++tq) {
    int col = tile0 + tq * 16 + rw;
    float bi = bias ? bias[col] : 0.f;
    v8f acc = (tq == 0) ? acc0 : (tq == 1) ? acc1 : (tq == 2) ? acc2 : acc3;
#pragma unroll
    for (int r = 0; r < 8; ++r) {
      float v = acc[r] + bi;
      if (act) v = (v >= 0.f) ? v : 0.01f * v;
      out[(size_t)(r + 8 * hi) * N + col] = v;
    }
  }
}

// ---- Scan GEMM 1: t(P,M,B) = A0(P,M,N) @ y(P,N,B) ---------------------------
// grid.x = P_*8 (one wave per (p, 64-row m-block)); contraction over N.
// B operand (y) loaded once per k-step, shared by 4 m-tiles.
__global__ void k_Ay(const float* __restrict__ A0, const float* __restrict__ yv,
                     float* __restrict__ t) {
  int pid = blockIdx.x;
  int p = pid >> 3;
  int mt0 = (pid & 7) * 64;
  int lane = threadIdx.x & 31;
  int rw = lane & 15, hi = lane >> 4;
  v8f acc0 = {}, acc1 = {}, acc2 = {}, acc3 = {};
  const float* ar0 = A0 + ((size_t)p * M_ + mt0 + 0  + rw) * N_;
  const float* ar1 = A0 + ((size_t)p * M_ + mt0 + 16 + rw) * N_;
  const float* ar2 = A0 + ((size_t)p * M_ + mt0 + 32 + rw) * N_;
  const float* ar3 = A0 + ((size_t)p * M_ + mt0 + 48 + rw) * N_;
  const float* yp = yv + (size_t)p * N_ * B_;
  for (int kb = 0; kb < N_; kb += 4) {
    int k0 = kb + 2 * hi;
    v2f b; b.x = yp[(size_t)k0 * B_ + rw]; b.y = yp[(size_t)(k0 + 1) * B_ + rw];
    v2f a;
    a.x = ar0[k0]; a.y = ar0[k0 + 1]; acc0 = wmma4(a, b, acc0);
    a.x = ar1[k0]; a.y = ar1[k0 + 1]; acc1 = wmma4(a, b, acc1);
    a.x = ar2[k0]; a.y = ar2[k0 + 1]; acc2 = wmma4(a, b, acc2);
    a.x = ar3[k0]; a.y = ar3[k0 + 1]; acc3 = wmma4(a, b, acc3);
  }
  float* tp = t + ((size_t)p * M_ + mt0) * B_;
#pragma unroll
  for (int r = 0; r < 8; ++r) {
    tp[(size_t)(r + 8 * hi) * B_ + rw]      = acc0[r];
    tp[(size_t)(16 + r + 8 * hi) * B_ + rw] = acc1[r];
    tp[(size_t)(32 + r + 8 * hi) * B_ + rw] = acc2[r];
    tp[(size_t)(48 + r + 8 * hi) * B_ + rw] = acc3[r];
  }
}

// ---- Scan GEMM 2: out(P,N,B) = A0^T(P,N,M) @ src(P,M,B) ---------------------
// grid.x = P_*16 (one wave per (p, 64-row n-block)); contraction over M.
// B operand (src) loaded once per k-step, shared by 4 n-tiles; A^T operand
// reads A0[m][n] with n = lane%16 -> lane-contiguous loads.
__global__ void k_AtX(const float* __restrict__ A0, const float* __restrict__ src,
                      float* __restrict__ out) {
  int pid = blockIdx.x;
  int p = pid >> 4;
  int nt0 = (pid & 15) * 64;
  int lane = threadIdx.x & 31;
  int ln = lane & 15, hi = lane >> 4;
  v8f acc0 = {}, acc1 = {}, acc2 = {}, acc3 = {};
  const float* Ap = A0 + (size_t)p * M_ * N_ + nt0 + ln;
  const float* sp = src + (size_t)p * M_ * B_;
  for (int kb = 0; kb < M_; kb += 4) {
    int k0 = kb + 2 * hi;
    v2f b; b.x = sp[(size_t)k0 * B_ + ln]; b.y = sp[(size_t)(k0 + 1) * B_ + ln];
    const float* a0p = Ap + (size_t)k0 * N_;
    const float* a1p = Ap + (size_t)(k0 + 1) * N_;
    v2f a;
    a.x = a0p[0];  a.y = a1p[0];  acc0 = wmma4(a, b, acc0);
    a.x = a0p[16]; a.y = a1p[16]; acc1 = wmma4(a, b, acc1);
    a.x = a0p[32]; a.y = a1p[32]; acc2 = wmma4(a, b, acc2);
    a.x = a0p[48]; a.y = a1p[48]; acc3 = wmma4(a, b, acc3);
  }
  float* op = out + ((size_t)p * N_ + nt0) * B_;
#pragma unroll
  for (int r = 0; r < 8; ++r) {
    op[(size_t)(r + 8 * hi) * B_ + ln]      = acc0[r];
    op[(size_t)(16 + r + 8 * hi) * B_ + ln] = acc1[r];
    op[(size_t)(32 + r + 8 * hi) * B_ + ln] = acc2[r];
    op[(size_t)(48 + r + 8 * hi) * B_ + ln] = acc3[r];
  }
}

// ---------------- Small helper kernels ---------------------------------------
__global__ void k_zero(float* __restrict__ p, int n) {
  int i = blockIdx.x * blockDim.x + threadIdx.x;
  if (i < n) p[i] = 0.f;
}

// (B,P,X) -> (P,X,B)
__global__ void k_tr_bpx(const float* __restrict__ in, float* __restrict__ out, int X) {
  int i = blockIdx.x * blockDim.x + threadIdx.x;
  int total = P_ * X * B_;
  if (i >= total) return;
  int b = i % B_;
  int x = (i / B_) % X;
  int p = i / (B_ * X);
  out[i] = in[((size_t)b * P_ + p) * X + x];
}

// deg[b][p] += #edges dst==p ; snb[b][p] += #edges src==p  (integer-exact f32)
__global__ void k_counts(const int* __restrict__ ei, float* __restrict__ deg,
                         float* __restrict__ snb) {
  int i = blockIdx.x * blockDim.x + threadIdx.x;
  if (i >= B_ * E_) return;
  int b = i / E_, e = i % E_;
  int s = ei[b * 2 * E_ + e];
  int d = ei[b * 2 * E_ + E_ + e];
  atomicAdd(&deg[b * P_ + d], 1.0f);
  atomicAdd(&snb[b * P_ + s], 1.0f);
}

// h(B,P,Fo) = x(B,P,Fi) @ W(Fi,Fo)
__global__ void k_mm_bp(const float* __restrict__ x, const float* __restrict__ W,
                        float* __restrict__ h, int Fi, int Fo) {
  int i = blockIdx.x * blockDim.x + threadIdx.x;
  int total = B_ * P_ * Fo;
  if (i >= total) return;
  int j = i % Fo;
  int p = (i / Fo) % P_;
  int b = i / (Fo * P_);
  const float* xr = x + ((size_t)b * P_ + p) * Fi;
  float acc = 0.f;
  for (int c = 0; c < Fi; ++c) acc += xr[c] * W[(size_t)c * Fo + j];
  h[i] = acc;
}

// out = h * (1/deg) + bias   (dinv^2 self term)
__global__ void k_gcn_init(const float* __restrict__ h, const float* __restrict__ deg,
                           const float* __restrict__ bias, float* __restrict__ out,
                           int Fo) {
  int i = blockIdx.x * blockDim.x + threadIdx.x;
  int total = B_ * P_ * Fo;
  if (i >= total) return;
  int j = i % Fo;
  int p = (i / Fo) % P_;
  int b = i / (Fo * P_);
  float dv = deg[b * P_ + p] + 1.f;
  float di = rsqrtf(dv);
  out[i] = h[i] * di * di + bias[j];
}

// out[b][d] += h[b][s] * dinv[s]*dinv[d]  over edges
__global__ void k_gcn_scat(const float* __restrict__ h, const float* __restrict__ deg,
                           const int* __restrict__ ei, float* __restrict__ out, int Fo) {
  int i = blockIdx.x * blockDim.x + threadIdx.x;
  int total = B_ * E_ * Fo;
  if (i >= total) return;
  int j = i % Fo;
  int e = (i / Fo) % E_;
  int b = i / (Fo * E_);
  int s = ei[b * 2 * E_ + e];
  int d = ei[b * 2 * E_ + E_ + e];
  float w = rsqrtf(deg[b * P_ + s] + 1.f) * rsqrtf(deg[b * P_ + d] + 1.f);
  atomicAdd(&out[((size_t)b * P_ + d) * Fo + j], h[((size_t)b * P_ + s) * Fo + j] * w);
}

__global__ void k_lrelu(float* __restrict__ p, int n) {
  int i = blockIdx.x * blockDim.x + threadIdx.x;
  if (i >= n) return;
  float v = p[i];
  p[i] = (v >= 0.f) ? v : 0.01f * v;
}

// pooled(B,128) = mean over P of x(B,P,128)
__global__ void k_pool(const float* __restrict__ x, float* __restrict__ pooled) {
  int i = blockIdx.x * blockDim.x + threadIdx.x;
  if (i >= B_ * 128) return;
  int j = i % 128, b = i / 128;
  float acc = 0.f;
  for (int p = 0; p < P_; ++p) acc += x[((size_t)b * P_ + p) * 128 + j];
  pooled[i] = acc * (1.0f / P_);
}

// f1o(B,64) = lrelu(pooled @ W_f1 + b_f1)
__global__ void k_f1(const float* __restrict__ x, const float* __restrict__ W,
                     const float* __restrict__ bias, float* __restrict__ o) {
  int i = blockIdx.x * blockDim.x + threadIdx.x;
  if (i >= B_ * 64) return;
  int j = i % 64, b = i / 64;
  float acc = bias[j];
  for (int c = 0; c < 128; ++c) acc += x[b * 128 + c] * W[c * 64 + j];
  o[i] = (acc >= 0.f) ? acc : 0.01f * acc;
}

// raw(B,80) = f1o @ W_f2 + b_f2
__global__ void k_raw(const float* __restrict__ x, const float* __restrict__ W,
                      const float* __restrict__ bias, float* __restrict__ o) {
  int i = blockIdx.x * blockDim.x + threadIdx.x;
  if (i >= B_ * 80) return;
  int j = i % 80, b = i / 80;
  float acc = bias[j];
  for (int c = 0; c < 64; ++c) acc += x[b * 64 + c] * W[c * 80 + j];
  o[i] = acc;
}

// hyp(B,KST,4) = sigmoid(cumsum_k raw) * max_param
__global__ void k_hyp(const float* __restrict__ raw, const float* __restrict__ maxp,
                      float* __restrict__ hyp) {
  int i = blockIdx.x * blockDim.x + threadIdx.x;
  if (i >= B_ * 4) return;
  int j = i % 4, b = i / 4;
  float c = 0.f;
  for (int k = 0; k < KST; ++k) {
    c += raw[b * (KST * 4) + k * 4 + j];
    float s = 1.0f / (1.0f + expf(-c));
    hyp[((size_t)b * KST + k) * 4 + j] = s * maxp[j];
  }
}

// y_{k+1} = y - alpha*(AtAy - Atb + sign(y)*tau + U*sum_nb + dlt*rho); emit Y[k]
__global__ void k_step_y(const float* __restrict__ g0, const float* __restrict__ Atb,
                         const float* __restrict__ snb, float* __restrict__ y,
                         const float* __restrict__ u, const float* __restrict__ dlt,
                         const float* __restrict__ hyp, int k, float* __restrict__ Yout) {
  int i = blockIdx.x * blockDim.x + threadIdx.x;
  int total = P_ * N_ * B_;
  if (i >= total) return;
  int b = i % B_;
  int n = (i / B_) % N_;
  int p = i / (B_ * N_);
  const float* hk = hyp + ((size_t)b * KST + k) * 4;
  float alpha = hk[0], tau = hk[1], rho = hk[2];
  float yv = y[i];
  float sg = (yv > 0.f) ? 1.f : ((yv < 0.f) ? -1.f : 0.f);
  float g = g0[i] - Atb[i] + sg * tau + u[i] * snb[b * P_ + p] + dlt[i] * rho;
  float yn = yv - alpha * g;
  y[i] = yn;
  Yout[(((size_t)k * B_ + b) * P_ + p) * N_ + n] = yn;
}

// dlt[s] += (y[s]-y[d]); dlt[d] -= (y[s]-y[d])  over edges (y in (P,N,B))
__global__ void k_delta(const float* __restrict__ y, const int* __restrict__ ei,
                        float* __restrict__ dlt) {
  int i = blockIdx.x * blockDim.x + threadIdx.x;
  int total = E_ * N_ * B_;
  if (i >= total) return;
  int b = i % B_;
  int n = (i / B_) % N_;
  int e = i / (B_ * N_);
  int s = ei[b * 2 * E_ + e];
  int d = ei[b * 2 * E_ + E_ + e];
  float diff = y[((size_t)s * N_ + n) * B_ + b] - y[((size_t)d * N_ + n) * B_ + b];
  atomicAdd(&dlt[((size_t)s * N_ + n) * B_ + b], diff);
  atomicAdd(&dlt[((size_t)d * N_ + n) * B_ + b], -diff);
}

// U += dlt * eta
__global__ void k_step_u(float* __restrict__ u, const float* __restrict__ dlt,
                         const float* __restrict__ hyp, int k) {
  int i = blockIdx.x * blockDim.x + threadIdx.x;
  int total = P_ * N_ * B_;
  if (i >= total) return;
  int b = i % B_;
  u[i] += dlt[i] * hyp[((size_t)b * KST + k) * 4 + 3];
}

static inline int cdiv(int a, int b) { return (a + b - 1) / b; }

extern "C" void kernel_launch(void* const* d_in, const int* in_sizes, int n_in,
                              void* d_out, int out_size, void* d_ws, size_t ws_size,
                              hipStream_t stream) {
  (void)in_sizes; (void)n_in; (void)out_size; (void)ws_size;
  const float* b_in = (const float*)d_in[0];   // (B,P,M)
  const float* A0   = (const float*)d_in[1];   // (P,M,N)
  const int*   ei   = (const int*)d_in[2];     // (B,2,400)
  const float* maxp = (const float*)d_in[3];   // (4)
  const float* W_e1 = (const float*)d_in[4];  const float* b_e1 = (const float*)d_in[5];
  const float* W_e2 = (const float*)d_in[6];  const float* b_e2 = (const float*)d_in[7];
  const float* W_e3 = (const float*)d_in[8];  const float* b_e3 = (const float*)d_in[9];
  const float* W_c1 = (const float*)d_in[10]; const float* b_c1 = (const float*)d_in[11];
  const float* W_c2 = (const float*)d_in[12]; const float* b_c2 = (const float*)d_in[13];
  const float* W_f1 = (const float*)d_in[14]; const float* b_f1 = (const float*)d_in[15];
  const float* W_f2 = (const float*)d_in[16]; const float* b_f2 = (const float*)d_in[17];
  const float* y0 = (const float*)d_in[18];
  const float* u0 = (const float*)d_in[19];
  const float* d0 = (const float*)d_in[20];
  float* out = (float*)d_out;

  // Workspace carve (floats); total ~22.8 MB
  float* wp = (float*)d_ws;
  float* x1  = wp; wp += (size_t)B_ * 3200;
  float* x2  = wp; wp += (size_t)B_ * 6400;
  float* x3  = wp; wp += (size_t)B_ * 12800;
  float* h1  = wp; wp += (size_t)B_ * P_ * 128;
  float* g1  = wp; wp += (size_t)B_ * P_ * 128;
  float* h2  = wp; wp += (size_t)B_ * P_ * 128;
  float* g2  = wp; wp += (size_t)B_ * P_ * 128;
  float* deg = wp; wp += B_ * P_;
  float* snb = wp; wp += B_ * P_;
  float* pooled = wp; wp += B_ * 128;
  float* f1o = wp; wp += B_ * 64;
  float* raw = wp; wp += B_ * 80;
  float* hyp = wp; wp += B_ * KST * 4;
  float* bt  = wp; wp += (size_t)P_ * M_ * B_;   // b transposed (P,M,B)
  float* Atb = wp; wp += (size_t)P_ * N_ * B_;
  float* yv  = wp; wp += (size_t)P_ * N_ * B_;
  float* uv  = wp; wp += (size_t)P_ * N_ * B_;
  float* dv  = wp; wp += (size_t)P_ * N_ * B_;
  float* tt  = wp; wp += (size_t)P_ * M_ * B_;   // t = A*y (P,M,B)
  float* g0  = wp; wp += (size_t)P_ * N_ * B_;

  const int T = 256;

  // ---- Encoder MLP (WMMA, 16-row GEMMs; 64 cols per wave) ----
  k_gemm16<<<3200 / 64, 32, 0, stream>>>(b_in, W_e1, b_e1, x1, 25600, 3200, 1);
  k_gemm16<<<6400 / 64, 32, 0, stream>>>(x1, W_e2, b_e2, x2, 3200, 6400, 1);
  k_gemm16<<<12800 / 64, 32, 0, stream>>>(x2, W_e3, b_e3, x3, 6400, 12800, 0);

  // ---- Graph degree / src counts ----
  k_zero<<<cdiv(B_ * P_, T), T, 0, stream>>>(deg, B_ * P_);
  k_zero<<<cdiv(B_ * P_, T), T, 0, stream>>>(snb, B_ * P_);
  k_counts<<<cdiv(B_ * E_, T), T, 0, stream>>>(ei, deg, snb);

  // ---- GCN layer 1 (256 -> 128) ----
  int gcn_n = B_ * P_ * 128;
  k_mm_bp<<<cdiv(gcn_n, T), T, 0, stream>>>(x3, W_c1, h1, 256, 128);
  k_gcn_init<<<cdiv(gcn_n, T), T, 0, stream>>>(h1, deg, b_c1, g1, 128);
  k_gcn_scat<<<cdiv(B_ * E_ * 128, T), T, 0, stream>>>(h1, deg, ei, g1, 128);
  k_lrelu<<<cdiv(gcn_n, T), T, 0, stream>>>(g1, gcn_n);

  // ---- GCN layer 2 (128 -> 128) ----
  k_mm_bp<<<cdiv(gcn_n, T), T, 0, stream>>>(g1, W_c2, h2, 128, 128);
  k_gcn_init<<<cdiv(gcn_n, T), T, 0, stream>>>(h2, deg, b_c2, g2, 128);
  k_gcn_scat<<<cdiv(B_ * E_ * 128, T), T, 0, stream>>>(h2, deg, ei, g2, 128);
  k_lrelu<<<cdiv(gcn_n, T), T, 0, stream>>>(g2, gcn_n);

  // ---- Head: mean-pool, f1, f2, cumsum+sigmoid ----
  k_pool<<<cdiv(B_ * 128, T), T, 0, stream>>>(g2, pooled);
  k_f1<<<cdiv(B_ * 64, T), T, 0, stream>>>(pooled, W_f1, b_f1, f1o);
  k_raw<<<cdiv(B_ * 80, T), T, 0, stream>>>(f1o, W_f2, b_f2, raw);
  k_hyp<<<1, 64, 0, stream>>>(raw, maxp, hyp);

  // ---- Atb = A^T b (WMMA), scan state transposed to (P,N,B) ----
  k_tr_bpx<<<cdiv(P_ * M_ * B_, T), T, 0, stream>>>(b_in, bt, M_);
  k_AtX<<<P_ * 16, 32, 0, stream>>>(A0, bt, Atb);
  k_tr_bpx<<<cdiv(P_ * N_ * B_, T), T, 0, stream>>>(y0, yv, N_);
  k_tr_bpx<<<cdiv(P_ * N_ * B_, T), T, 0, stream>>>(u0, uv, N_);
  k_tr_bpx<<<cdiv(P_ * N_ * B_, T), T, 0, stream>>>(d0, dv, N_);

  // ---- Scan: 20 steps, AtAy computed as A^T (A y) (A stays L2-resident) ----
  int ew_n = P_ * N_ * B_;
  for (int k = 0; k < KST; ++k) {
    k_Ay<<<P_ * 8, 32, 0, stream>>>(A0, yv, tt);
    k_AtX<<<P_ * 16, 32, 0, stream>>>(A0, tt, g0);
    k_step_y<<<cdiv(ew_n, T), T, 0, stream>>>(g0, Atb, snb, yv, uv, dv, hyp, k, out);
    k_zero<<<cdiv(ew_n, T), T, 0, stream>>>(dv, ew_n);
    k_delta<<<cdiv(E_ * N_ * B_, T), T, 0, stream>>>(yv, ei, dv);
    k_step_u<<<cdiv(ew_n, T), T, 0, stream>>>(uv, dv, hyp, k);
  }
}